// EQMixer_40106404610355
// MI455X (gfx1250) — compile-verified
//
#include <hip/hip_runtime.h>
#include <math.h>

// EQMixer on MI455X (gfx1250, wave32, WMMA, async-to-LDS staging).
// Masked-state mix == full pre-activation minus a rank-64 delta GEMM per
// agent => 2x base GEMM cost instead of 33x. All GEMMs use
// V_WMMA_F32_16X16X4_F32 (fp32-faithful). W tiles are staged into LDS once
// per 4-wave workgroup (async double-buffered via GLOBAL_LOAD_ASYNC_TO_LDS_B128
// + s_wait_asynccnt when the toolchain exposes the builtins).

typedef __attribute__((ext_vector_type(2))) float v2f;
typedef __attribute__((ext_vector_type(8))) float v8f;

// GCC-style vector to match the async builtin's parameter type exactly:
//   int __attribute__((vector_size(16))) __device__ *   (AS1, global)
typedef int v4i_vs __attribute__((vector_size(16)));
typedef __attribute__((address_space(1))) v4i_vs* as1_v4i_ptr;
typedef __attribute__((address_space(3))) v4i_vs* as3_v4i_ptr;

#define B_ROWS 2048   // BS * T
#define SDIM   2048   // STATE_DIM
#define NAG    32     // N_AGENTS
#define EMB    64     // EMBED

#define KC       64   // K rows staged per chunk
#define WSTRIDE  68   // LDS row stride in floats (16B aligned, bank-skewed)
#define STAGE_OPS 8   // async-LDS instructions per wave per stage

#if defined(__has_builtin)
#if __has_builtin(__builtin_amdgcn_global_load_async_to_lds_b128)
#define HAVE_ASYNC_LDS 1
#endif
#endif
#ifndef HAVE_ASYNC_LDS
#define HAVE_ASYNC_LDS 0
#endif

static __device__ inline v8f wmma4(v2f a, v2f b, v8f c) {
  // D = A(16x4) * B(4x16) + C, fp32.
  return __builtin_amdgcn_wmma_f32_16x16x4_f32(false, a, false, b, (short)0, c,
                                               false, false);
}

// Stage a 64(K) x 64(N) fp32 W block into LDS (row stride WSTRIDE floats).
// 128 threads, 8 x b128 per thread. Async path uses ASYNCcnt.
static __device__ inline void stage_chunk(const float* __restrict__ Wt, int ldw,
                                          float* __restrict__ buf, int tid) {
#pragma unroll
  for (int j = 0; j < STAGE_OPS; ++j) {
    const int idx4 = tid + j * 128;      // 1024 float4s
    const int row = idx4 >> 4;           // 16 float4s per row
    const int col = (idx4 & 15) * 4;
#if HAVE_ASYNC_LDS
    __builtin_amdgcn_global_load_async_to_lds_b128(
        (as1_v4i_ptr)(Wt + (size_t)row * ldw + col),
        (as3_v4i_ptr)(buf + row * WSTRIDE + col),
        0, 0);
#else
    const float4 v = *(const float4*)(Wt + (size_t)row * ldw + col);
    *(float4*)(buf + row * WSTRIDE + col) = v;
#endif
  }
}

static __device__ inline void wait_stage_pending() {  // allow 1 stage in flight
#if HAVE_ASYNC_LDS
#if __has_builtin(__builtin_amdgcn_s_wait_asynccnt)
  __builtin_amdgcn_s_wait_asynccnt(STAGE_OPS);
#else
  asm volatile("s_wait_asynccnt %0" ::"i"(STAGE_OPS));
#endif
#endif
}
static __device__ inline void wait_stage_all() {
#if HAVE_ASYNC_LDS
#if __has_builtin(__builtin_amdgcn_s_wait_asynccnt)
  __builtin_amdgcn_s_wait_asynccnt(0);
#else
  asm volatile("s_wait_asynccnt 0");
#endif
#endif
}

// ---------------------------------------------------------------------------
// Kernel 1: P = st @ W + bias. 4 waves/WG; wave = 32 rows (2 M-frags) x 64
// cols; W chunk shared via LDS, double-buffered.
// ---------------------------------------------------------------------------
__global__ __launch_bounds__(128)
void eqmix_gemm(const float* __restrict__ A, const float* __restrict__ W,
                const float* __restrict__ bias, float* __restrict__ P, int N) {
  __shared__ float wbuf[2][KC * WSTRIDE];
  const int tid = threadIdx.x;
  const int lane = tid & 31, wave = tid >> 5;
  const int n0 = blockIdx.x * 64;
  const int m0 = blockIdx.y * 128 + wave * 32;
  const int half = lane >> 4, l15 = lane & 15;

  const v8f zero = {0.f, 0.f, 0.f, 0.f, 0.f, 0.f, 0.f, 0.f};
  v8f acc[2][4] = {{zero, zero, zero, zero}, {zero, zero, zero, zero}};

  const float* Arow0 = A + (size_t)(m0 + l15) * SDIM;
  const float* Arow1 = A + (size_t)(m0 + 16 + l15) * SDIM;

  stage_chunk(W + n0, N, wbuf[0], tid);
  const int NCH = SDIM / KC;
  for (int c = 0; c < NCH; ++c) {
    const int cur = c & 1;
    if (c + 1 < NCH) {
      stage_chunk(W + (size_t)(c + 1) * KC * N + n0, N, wbuf[cur ^ 1], tid);
      wait_stage_pending();
    } else {
      wait_stage_all();
    }
    __syncthreads();
    const float* buf = wbuf[cur];
    const int kb = c * KC;
#pragma unroll 4
    for (int ks = 0; ks < 16; ++ks) {
      const int kk = ks * 4 + half * 2;
      v2f a0, a1;
      a0.x = Arow0[kb + kk];
      a0.y = Arow0[kb + kk + 1];
      a1.x = Arow1[kb + kk];
      a1.y = Arow1[kb + kk + 1];
#pragma unroll
      for (int n = 0; n < 4; ++n) {
        v2f b;
        b.x = buf[kk * WSTRIDE + n * 16 + l15];
        b.y = buf[(kk + 1) * WSTRIDE + n * 16 + l15];
        acc[0][n] = wmma4(a0, b, acc[0][n]);
        acc[1][n] = wmma4(a1, b, acc[1][n]);
      }
    }
    __syncthreads();
  }

  // C layout: VGPR r holds M = r (lanes 0-15) / r+8 (lanes 16-31), N = lane&15
#pragma unroll
  for (int f = 0; f < 2; ++f) {
#pragma unroll
    for (int n = 0; n < 4; ++n) {
      const int col = n0 + n * 16 + l15;
      const float bcol = bias[col];
#pragma unroll
      for (int r = 0; r < 8; ++r) {
        const int m = m0 + f * 16 + r + half * 8;
        P[(size_t)m * N + col] = acc[f][n][r] + bcol;
      }
    }
  }
}

// ---------------------------------------------------------------------------
// Kernel 2: q_tot[b] from full pre-activations. 64 threads (thread = embed e).
// ---------------------------------------------------------------------------
__global__ __launch_bounds__(64)
void eqmix_qtot(const float* __restrict__ P1, const float* __restrict__ Pf,
                const float* __restrict__ Pb, const float* __restrict__ Pv,
                const float* __restrict__ qs, const float* __restrict__ Wv2,
                const float* __restrict__ bv2, float* __restrict__ q_tot) {
  const int b = blockIdx.x;
  const int e = threadIdx.x;
  __shared__ float qsh[NAG];
  __shared__ float red[2];
  if (e < NAG) qsh[e] = qs[(size_t)b * NAG + e];
  __syncthreads();

  const float* p = P1 + (size_t)b * SDIM;
  float acc = 0.f;
#pragma unroll 8
  for (int a = 0; a < NAG; ++a) acc += qsh[a] * fabsf(p[a * EMB + e]);

  float h = acc + Pb[(size_t)b * EMB + e];
  h = h > 0.f ? h : (__expf(h) - 1.f);                 // elu
  float contrib = h * fabsf(Pf[(size_t)b * EMB + e]);  // hidden . |w_final|
  float pv = Pv[(size_t)b * EMB + e];
  contrib += fmaxf(pv, 0.f) * Wv2[e];                  // V(s)

  for (int off = 16; off > 0; off >>= 1) contrib += __shfl_down(contrib, off, 32);
  if ((e & 31) == 0) red[e >> 5] = contrib;
  __syncthreads();
  if (e == 0) q_tot[b] = red[0] + red[1] + bv2[0];
}

// ---------------------------------------------------------------------------
// Kernel 3: counterfactual q1[i, b]. grid = (B/64, NAG), 4 waves/WG.
// Wave = 16 rows. A chunk (st columns i*64..) preloaded in registers and
// reused across all 35 delta GEMMs; W blocks staged via LDS double buffer.
// Chunk order: t=0..31 -> Ww1 column blocks (hidden accum),
//              t=32 -> Wb1 slice (elu), t=33 -> Wwf slice (dot),
//              t=34 -> Wv1 slice (V(s)).
// ---------------------------------------------------------------------------
__global__ __launch_bounds__(128)
void eqmix_q1(const float* __restrict__ st, const float* __restrict__ P1,
              const float* __restrict__ Pf, const float* __restrict__ Pb,
              const float* __restrict__ Pv, const float* __restrict__ qs,
              const float* __restrict__ Ww1, const float* __restrict__ Wwf,
              const float* __restrict__ Wb1, const float* __restrict__ Wv1,
              const float* __restrict__ Wv2, const float* __restrict__ bv2,
              float* __restrict__ q1) {
  __shared__ float wbuf[2][KC * WSTRIDE];
  __shared__ float qsh[64 * NAG];
  const int tid = threadIdx.x;
  const int lane = tid & 31, wave = tid >> 5;
  const int m0 = blockIdx.x * 64;        // 64 rows per WG
  const int i = blockIdx.y;              // masked agent
  const int half = lane >> 4, l15 = lane & 15;
  const int mw = m0 + wave * 16;         // this wave's 16 rows

  for (int t = tid; t < 64 * NAG; t += 128)
    qsh[t] = qs[(size_t)(m0 + (t >> 5)) * NAG + (t & 31)];

  // Preload this wave's 16x64 A chunk as 16 k-step fragments (reused 35x).
  v2f av[16];
  {
    const float* Arow = st + (size_t)(mw + l15) * SDIM + i * EMB;
#pragma unroll
    for (int ks = 0; ks < 16; ++ks) {
      const int kk = ks * 4 + half * 2;
      av[ks].x = Arow[kk];
      av[ks].y = Arow[kk + 1];
    }
  }

  const float* Ww1s = Ww1 + (size_t)i * EMB * SDIM;
  const float* tailW[3] = {Wb1 + (size_t)i * EMB * EMB,
                           Wwf + (size_t)i * EMB * EMB,
                           Wv1 + (size_t)i * EMB * EMB};

  const v8f zero = {0.f, 0.f, 0.f, 0.f, 0.f, 0.f, 0.f, 0.f};
  v8f hid[4] = {zero, zero, zero, zero};
  float yacc[8];
#pragma unroll
  for (int r = 0; r < 8; ++r) yacc[r] = 0.f;

  stage_chunk(Ww1s, SDIM, wbuf[0], tid);
  for (int t = 0; t < 35; ++t) {
    const int cur = t & 1;
    if (t + 1 < 35) {
      const float* Wn;
      int ldw;
      if (t + 1 < 32) { Wn = Ww1s + (size_t)(t + 1) * EMB; ldw = SDIM; }
      else           { Wn = tailW[t + 1 - 32];             ldw = EMB;  }
      stage_chunk(Wn, ldw, wbuf[cur ^ 1], tid);
      wait_stage_pending();
    } else {
      wait_stage_all();
    }
    __syncthreads();

    const float* buf = wbuf[cur];
    v8f d[4] = {zero, zero, zero, zero};
#pragma unroll 4
    for (int ks = 0; ks < 16; ++ks) {
      const int kk = ks * 4 + half * 2;
#pragma unroll
      for (int n = 0; n < 4; ++n) {
        v2f b;
        b.x = buf[kk * WSTRIDE + n * 16 + l15];
        b.y = buf[(kk + 1) * WSTRIDE + n * 16 + l15];
        d[n] = wmma4(av[ks], b, d[n]);
      }
    }

    if (t < 32) {                        // hidden += qs_a * |P1 - delta|
#pragma unroll
      for (int n = 0; n < 4; ++n) {
        const int col = n * 16 + l15;
#pragma unroll
        for (int r = 0; r < 8; ++r) {
          const int m = r + half * 8;
          const float p = P1[(size_t)(mw + m) * SDIM + t * EMB + col];
          hid[n][r] += qsh[(wave * 16 + m) * NAG + t] * fabsf(p - d[n][r]);
        }
      }
    } else if (t == 32) {                // hid = elu(hid + (Pb - db))
#pragma unroll
      for (int n = 0; n < 4; ++n) {
        const int col = n * 16 + l15;
#pragma unroll
        for (int r = 0; r < 8; ++r) {
          const int m = mw + r + half * 8;
          float h = hid[n][r] + (Pb[(size_t)m * EMB + col] - d[n][r]);
          hid[n][r] = h > 0.f ? h : (__expf(h) - 1.f);
        }
      }
    } else if (t == 33) {                // yacc += hid * |Pf - df|
#pragma unroll
      for (int n = 0; n < 4; ++n) {
        const int col = n * 16 + l15;
#pragma unroll
        for (int r = 0; r < 8; ++r) {
          const int m = mw + r + half * 8;
          yacc[r] += hid[n][r] * fabsf(Pf[(size_t)m * EMB + col] - d[n][r]);
        }
      }
    } else {                             // yacc += relu(Pv - dv) * Wv2
#pragma unroll
      for (int n = 0; n < 4; ++n) {
        const int col = n * 16 + l15;
        const float wv2c = Wv2[col];
#pragma unroll
        for (int r = 0; r < 8; ++r) {
          const int m = mw + r + half * 8;
          const float pv = Pv[(size_t)m * EMB + col] - d[n][r];
          yacc[r] += fmaxf(pv, 0.f) * wv2c;
        }
      }
    }
    __syncthreads();
  }

  // reduce over the 16 lanes of each half (xor<=8 stays inside each half)
#pragma unroll
  for (int r = 0; r < 8; ++r) {
    float s = yacc[r];
    s += __shfl_xor(s, 1, 32);
    s += __shfl_xor(s, 2, 32);
    s += __shfl_xor(s, 4, 32);
    s += __shfl_xor(s, 8, 32);
    if (l15 == 0) {
      const int m = mw + r + half * 8;
      q1[(size_t)i * B_ROWS + m] = s + bv2[0];
    }
  }
}

// ---------------------------------------------------------------------------
// Kernel 4: wc = normalize(|q_tot - q1|); final mix with qs*wc -> out[b]
// ---------------------------------------------------------------------------
__global__ __launch_bounds__(64)
void eqmix_final(const float* __restrict__ P1, const float* __restrict__ Pf,
                 const float* __restrict__ Pb, const float* __restrict__ Pv,
                 const float* __restrict__ qs, const float* __restrict__ Wv2,
                 const float* __restrict__ bv2, const float* __restrict__ q_tot,
                 const float* __restrict__ q1, float* __restrict__ out) {
  const int b = blockIdx.x;
  const int e = threadIdx.x;
  __shared__ float qsh[NAG];
  __shared__ float red[2];

  if (e < NAG) {  // wave 0: build wc and reweight qs
    const float w = fabsf(q_tot[b] - q1[(size_t)e * B_ROWS + b]);
    float s = w * w;
    s += __shfl_xor(s, 1, 32);
    s += __shfl_xor(s, 2, 32);
    s += __shfl_xor(s, 4, 32);
    s += __shfl_xor(s, 8, 32);
    s += __shfl_xor(s, 16, 32);
    const float norm = fmaxf(sqrtf(s), 1e-12f);
    qsh[e] = qs[(size_t)b * NAG + e] * (w / norm);
  }
  __syncthreads();

  const float* p = P1 + (size_t)b * SDIM;
  float acc = 0.f;
#pragma unroll 8
  for (int a = 0; a < NAG; ++a) acc += qsh[a] * fabsf(p[a * EMB + e]);

  float h = acc + Pb[(size_t)b * EMB + e];
  h = h > 0.f ? h : (__expf(h) - 1.f);
  float contrib = h * fabsf(Pf[(size_t)b * EMB + e]);
  float pv = Pv[(size_t)b * EMB + e];
  contrib += fmaxf(pv, 0.f) * Wv2[e];

  for (int off = 16; off > 0; off >>= 1) contrib += __shfl_down(contrib, off, 32);
  if ((e & 31) == 0) red[e >> 5] = contrib;
  __syncthreads();
  if (e == 0) out[b] = red[0] + red[1] + bv2[0];
}

// ---------------------------------------------------------------------------
extern "C" void kernel_launch(void* const* d_in, const int* in_sizes, int n_in,
                              void* d_out, int out_size, void* d_ws, size_t ws_size,
                              hipStream_t stream) {
  const float* qs  = (const float*)d_in[0];   // [B, 32]
  const float* st  = (const float*)d_in[1];   // [B, 2048]
  const float* Ww1 = (const float*)d_in[2];   // [2048, 2048]
  const float* bw1 = (const float*)d_in[3];   // [2048]
  const float* Wwf = (const float*)d_in[4];   // [2048, 64]
  const float* bwf = (const float*)d_in[5];   // [64]
  const float* Wb1 = (const float*)d_in[6];   // [2048, 64]
  const float* bb1 = (const float*)d_in[7];   // [64]
  const float* Wv1 = (const float*)d_in[8];   // [2048, 64]
  const float* bv1 = (const float*)d_in[9];   // [64]
  const float* Wv2 = (const float*)d_in[10];  // [64]
  const float* bv2 = (const float*)d_in[11];  // [1]
  float* out = (float*)d_out;                 // [B] (= [64,32,1])

  float* ws = (float*)d_ws;
  float* P1   = ws;                                   // B x 2048
  float* Pf   = P1 + (size_t)B_ROWS * SDIM;           // B x 64
  float* Pb   = Pf + (size_t)B_ROWS * EMB;            // B x 64
  float* Pv   = Pb + (size_t)B_ROWS * EMB;            // B x 64
  float* qtot = Pv + (size_t)B_ROWS * EMB;            // B
  float* q1   = qtot + B_ROWS;                        // 32 x B

  eqmix_gemm<<<dim3(SDIM / 64, B_ROWS / 128), 128, 0, stream>>>(st, Ww1, bw1, P1, SDIM);
  eqmix_gemm<<<dim3(1, B_ROWS / 128), 128, 0, stream>>>(st, Wwf, bwf, Pf, EMB);
  eqmix_gemm<<<dim3(1, B_ROWS / 128), 128, 0, stream>>>(st, Wb1, bb1, Pb, EMB);
  eqmix_gemm<<<dim3(1, B_ROWS / 128), 128, 0, stream>>>(st, Wv1, bv1, Pv, EMB);

  eqmix_qtot<<<B_ROWS, 64, 0, stream>>>(P1, Pf, Pb, Pv, qs, Wv2, bv2, qtot);

  eqmix_q1<<<dim3(B_ROWS / 64, NAG), 128, 0, stream>>>(st, P1, Pf, Pb, Pv, qs,
                                                       Ww1, Wwf, Wb1, Wv1, Wv2, bv2, q1);

  eqmix_final<<<B_ROWS, 64, 0, stream>>>(P1, Pf, Pb, Pv, qs, Wv2, bv2, qtot, q1, out);
}